// DeformableTransformerDecoderLayer_19421842112726
// MI455X (gfx1250) — compile-verified
//
#include <hip/hip_runtime.h>

// ---------------- problem constants ----------------
#define BB   8
#define LQ   900
#define DD   256
#define NH   8
#define DH   32
#define NLEV 4
#define NPTS 4
#define DFFN 1024
#define LEN  13294           // 100*100 + 50*50 + 25*25 + 13*13

typedef __attribute__((ext_vector_type(2))) float v2f;
typedef __attribute__((ext_vector_type(8))) float v8f;
typedef int v4i __attribute__((vector_size(16)));    // int4, matches builtin signature

#define AS1 __attribute__((address_space(1)))
#define AS3 __attribute__((address_space(3)))

#if defined(__has_builtin)
#if __has_builtin(__builtin_amdgcn_global_load_async_to_lds_b128)
#define HAVE_ASYNC_LDS 1
#endif
#endif

__device__ __forceinline__ void wait_async0() {
#if defined(HAVE_ASYNC_LDS)
#if __has_builtin(__builtin_amdgcn_s_wait_asynccnt)
    __builtin_amdgcn_s_wait_asynccnt(0);
#else
    asm volatile("s_wait_asynccnt 0x0" ::: "memory");
#endif
#endif
}

#define LDS_STRIDE 72        // 64 + 8 pad: halves of a fragment hit disjoint banks

// Cooperatively stage a 64(K) x 64(N) weight chunk into LDS (async if available).
__device__ __forceinline__ void stage_panel(float* dst, const float* __restrict__ W,
                                            int k0, int ldw, int n0, int t)
{
    const int rb  = t >> 4;          // 0..15
    const int col = (t & 15) * 4;    // 0..60
    #pragma unroll
    for (int i = 0; i < 4; ++i) {
        const int row = rb + 16 * i;
        const float* g = W + (size_t)(k0 + row) * ldw + n0 + col;
        float* l = dst + row * LDS_STRIDE + col;
#if defined(HAVE_ASYNC_LDS)
        __builtin_amdgcn_global_load_async_to_lds_b128(
            (AS1 v4i*)const_cast<float*>(g), (AS3 v4i*)l, 0, 0);
#else
        *(float4*)l = *(const float4*)g;
#endif
    }
}

// ---------------------------------------------------------------------------
// fp32 WMMA GEMM:  C[M x N] = A[M x K] @ W[K x ldw (cols used: N)] + bias
// Block: 256 threads = 8 waves, each wave a 16x64 tile (4 accumulators).
// Weight panel double-buffered through LDS with async global->LDS copies.
// K stepped by 4 (V_WMMA_F32_16X16X4_F32). K must be a multiple of 64.
// ---------------------------------------------------------------------------
__global__ __launch_bounds__(256) void gemm_wmma_f32(
    const float* __restrict__ A, const float* __restrict__ W,
    const float* __restrict__ bias, float* __restrict__ C,
    int M, int K, int N, int ldw, int relu)
{
    __shared__ float ldsW[2][64 * LDS_STRIDE];

    const int t    = threadIdx.x;
    const int lane = t & 31;
    const int wave = t >> 5;
    const int m0 = (blockIdx.x * 8 + wave) * 16;
    const int n0 = blockIdx.y * 64;
    const bool active = (m0 < M);          // wave-uniform
    const int r    = lane & 15;
    const int half = lane >> 4;

    v8f acc[4] = {};
    const float* __restrict__ Arow = A + (size_t)(m0 + r) * K;

    const int nc = K >> 6;                 // chunks of 64 K-values
    stage_panel(&ldsW[0][0], W, 0, ldw, n0, t);

    for (int c = 0; c < nc; ++c) {
        wait_async0();
        __syncthreads();                   // staged chunk c visible to all waves
        if (c + 1 < nc)
            stage_panel(&ldsW[(c + 1) & 1][0], W, (c + 1) * 64, ldw, n0, t);

        if (active) {
            const float* buf = &ldsW[c & 1][0];
            const int k0 = c * 64;
            #pragma unroll 4
            for (int kk = 0; kk < 64; kk += 4) {
                const int ka = kk + 2 * half;
                v2f a;
                a.x = Arow[k0 + ka];
                a.y = Arow[k0 + ka + 1];
                const float* Wb = buf + ka * LDS_STRIDE + r;
                v2f b0, b1, b2, b3;
                b0.x = Wb[0];  b0.y = Wb[LDS_STRIDE];
                b1.x = Wb[16]; b1.y = Wb[LDS_STRIDE + 16];
                b2.x = Wb[32]; b2.y = Wb[LDS_STRIDE + 32];
                b3.x = Wb[48]; b3.y = Wb[LDS_STRIDE + 48];
                acc[0] = __builtin_amdgcn_wmma_f32_16x16x4_f32(false, a, false, b0, (short)0, acc[0], false, false);
                acc[1] = __builtin_amdgcn_wmma_f32_16x16x4_f32(false, a, false, b1, (short)0, acc[1], false, false);
                acc[2] = __builtin_amdgcn_wmma_f32_16x16x4_f32(false, a, false, b2, (short)0, acc[2], false, false);
                acc[3] = __builtin_amdgcn_wmma_f32_16x16x4_f32(false, a, false, b3, (short)0, acc[3], false, false);
            }
        }
        __syncthreads();                   // all reads of buf[c&1] done before restage
    }

    if (!active) return;
    #pragma unroll
    for (int j = 0; j < 4; ++j) {
        const int col = n0 + 16 * j + r;
        const float bv = bias ? bias[col] : 0.0f;
        #pragma unroll
        for (int v = 0; v < 8; ++v) {
            const int row = m0 + v + 8 * half;   // C/D layout: lanes16-31 hold M+8
            float val = acc[j][v] + bv;
            if (relu) val = fmaxf(val, 0.0f);
            C[(size_t)row * N + col] = val;
        }
    }
}

// ---------------------------------------------------------------------------
// Elementwise add: c = a + b
// ---------------------------------------------------------------------------
__global__ void add_vec(const float* __restrict__ a, const float* __restrict__ b,
                        float* __restrict__ c, int n)
{
    int i = blockIdx.x * blockDim.x + threadIdx.x;
    if (i < n) c[i] = a[i] + b[i];
}

// ---------------------------------------------------------------------------
// Fused residual + LayerNorm over rows of 256.  One wave per row (wave32).
// ---------------------------------------------------------------------------
__global__ __launch_bounds__(256) void add_ln256(
    const float* __restrict__ x, const float* __restrict__ res,
    const float* __restrict__ w, const float* __restrict__ bias,
    float* __restrict__ out, int rows)
{
    const int gw   = (int)((blockIdx.x * blockDim.x + threadIdx.x) >> 5);
    const int lane = threadIdx.x & 31;
    if (gw >= rows) return;
    const float* xr = x   + (size_t)gw * 256;
    const float* rr = res + (size_t)gw * 256;

    float v[8];
    float s = 0.0f;
    #pragma unroll
    for (int i = 0; i < 8; ++i) { v[i] = xr[lane * 8 + i] + rr[lane * 8 + i]; s += v[i]; }
    #pragma unroll
    for (int o = 16; o > 0; o >>= 1) s += __shfl_xor(s, o, 32);
    const float mean = s * (1.0f / 256.0f);

    float vs = 0.0f;
    #pragma unroll
    for (int i = 0; i < 8; ++i) { float d = v[i] - mean; vs += d * d; }
    #pragma unroll
    for (int o = 16; o > 0; o >>= 1) vs += __shfl_xor(vs, o, 32);
    const float rstd = rsqrtf(vs * (1.0f / 256.0f) + 1e-5f);

    #pragma unroll
    for (int i = 0; i < 8; ++i) {
        const int c = lane * 8 + i;
        out[(size_t)gw * 256 + c] = (v[i] - mean) * rstd * w[c] + bias[c];
    }
}

// ---------------------------------------------------------------------------
// Self-attention: one wave per (b,h,q), lane == head-dim channel (DH==32).
// Online-softmax over keys in chunks of 32 (no score matrix materialized).
// ---------------------------------------------------------------------------
__global__ __launch_bounds__(256) void self_attn(
    const float* __restrict__ qh, const float* __restrict__ kh,
    const float* __restrict__ vh, float* __restrict__ out)
{
    const int gw   = (int)((blockIdx.x * blockDim.x + threadIdx.x) >> 5);
    const int lane = threadIdx.x & 31;
    if (gw >= BB * NH * LQ) return;
    const int q = gw % LQ;
    const int h = (gw / LQ) % NH;
    const int b = gw / (LQ * NH);

    const float* qrow = qh + ((size_t)(b * LQ + q) * NH + h) * DH;
    float qv[32];
    #pragma unroll
    for (int d = 0; d < 32; ++d) qv[d] = qrow[d];

    const float scale = 0.17677669529663687f;   // 1/sqrt(32)
    float m = -3.0e38f, s = 0.0f, acc = 0.0f;

    for (int kc = 0; kc < LQ; kc += 32) {
        const int key = kc + lane;
        float sc = -3.0e38f;
        if (key < LQ) {
            const float* krow = kh + ((size_t)(b * LQ + key) * NH + h) * DH;
            float dot = 0.0f;
            #pragma unroll
            for (int d = 0; d < 32; ++d) dot += qv[d] * krow[d];
            sc = dot * scale;
        }
        float cm = sc;
        #pragma unroll
        for (int o = 16; o > 0; o >>= 1) cm = fmaxf(cm, __shfl_xor(cm, o, 32));
        const float mn   = fmaxf(m, cm);
        const float corr = __expf(m - mn);
        s *= corr; acc *= corr; m = mn;

        for (int j = 0; j < 32; ++j) {
            const int kj = kc + j;
            if (kj >= LQ) break;
            const float p = __expf(__shfl(sc, j, 32) - mn);
            s   += p;
            acc += p * vh[((size_t)(b * LQ + kj) * NH + h) * DH + lane];
        }
    }
    out[((size_t)(b * LQ + q) * NH + h) * DH + lane] = acc / s;
}

// ---------------------------------------------------------------------------
// MS-deformable sampling: one wave per (b,q,h), lane == channel.
// value (109MB) is L2-resident (192MB L2) so gathers stay on-chip.
// ---------------------------------------------------------------------------
__global__ __launch_bounds__(256) void msdeform_sample(
    const float* __restrict__ value, const float* __restrict__ off,
    const float* __restrict__ awlin, const float* __restrict__ ref,
    float* __restrict__ out)
{
    const int gw   = (int)((blockIdx.x * blockDim.x + threadIdx.x) >> 5);
    const int lane = threadIdx.x & 31;
    if (gw >= BB * NH * LQ) return;
    const int h = gw % NH;
    const int q = (gw / NH) % LQ;
    const int b = gw / (NH * LQ);

    const int Hs[4] = {100, 50, 25, 13};
    const int St[4] = {0, 10000, 12500, 13125};

    const float* ap = awlin + ((size_t)(b * LQ + q) * NH + h) * 16;
    float aws[16];
    float amax = -3.0e38f;
    #pragma unroll
    for (int i = 0; i < 16; ++i) { aws[i] = ap[i]; amax = fmaxf(amax, aws[i]); }
    float asum = 0.0f;
    #pragma unroll
    for (int i = 0; i < 16; ++i) { aws[i] = __expf(aws[i] - amax); asum += aws[i]; }
    const float inv = 1.0f / asum;

    const float* op = off + ((size_t)(b * LQ + q) * NH + h) * (NLEV * NPTS * 2);
    float acc = 0.0f;

    #pragma unroll
    for (int l = 0; l < NLEV; ++l) {
        const int   Wl = Hs[l], Hl = Hs[l];        // square levels
        const float refx = ref[((size_t)(b * LQ + q) * NLEV + l) * 2 + 0];
        const float refy = ref[((size_t)(b * LQ + q) * NLEV + l) * 2 + 1];
        const float* vbase = value + ((size_t)(b * LEN + St[l])) * DD + h * DH + lane;

        #pragma unroll
        for (int p = 0; p < NPTS; ++p) {
            const float ox = op[(l * NPTS + p) * 2 + 0];
            const float oy = op[(l * NPTS + p) * 2 + 1];
            const float lx = refx * (float)Wl + ox - 0.5f;   // (ref + off/W)*W - 0.5
            const float ly = refy * (float)Hl + oy - 0.5f;
            const float x0f = floorf(lx), y0f = floorf(ly);
            const float fx = lx - x0f, fy = ly - y0f;
            const int x0 = (int)x0f, y0 = (int)y0f;
            const float wlp = aws[l * NPTS + p] * inv;

            #pragma unroll
            for (int c = 0; c < 4; ++c) {
                const int dx = c & 1, dy = c >> 1;
                const int xi = x0 + dx, yi = y0 + dy;
                const float wf = (dx ? fx : 1.0f - fx) * (dy ? fy : 1.0f - fy);
                const bool valid = (xi >= 0) && (xi < Wl) && (yi >= 0) && (yi < Hl);
                const int xc = min(max(xi, 0), Wl - 1);
                const int yc = min(max(yi, 0), Hl - 1);
                const float g = vbase[(size_t)(yc * Wl + xc) * DD];
                acc += (valid ? wf : 0.0f) * wlp * g;
            }
        }
    }
    out[((size_t)(b * LQ + q) * NH + h) * DH + lane] = acc;
}

// ---------------------------------------------------------------------------
static inline void gemm(const float* A, const float* W, const float* bias, float* C,
                        int M, int K, int N, int ldw, int relu, hipStream_t s)
{
    dim3 grid((M / 16 + 7) / 8, N / 64);
    gemm_wmma_f32<<<grid, 256, 0, s>>>(A, W, bias, C, M, K, N, ldw, relu);
}

extern "C" void kernel_launch(void* const* d_in, const int* in_sizes, int n_in,
                              void* d_out, int out_size, void* d_ws, size_t ws_size,
                              hipStream_t stream)
{
    const float* tgt       = (const float*)d_in[0];
    const float* query_pos = (const float*)d_in[1];
    const float* refpts    = (const float*)d_in[2];
    const float* src       = (const float*)d_in[3];
    // d_in[4] spatial_shapes, d_in[5] level_start_index : compile-time constants
    const float* qkv_w = (const float*)d_in[6];
    const float* qkv_b = (const float*)d_in[7];
    const float* out_w = (const float*)d_in[8];
    const float* out_b = (const float*)d_in[9];
    const float* ln2_w = (const float*)d_in[10];
    const float* ln2_b = (const float*)d_in[11];
    const float* val_w = (const float*)d_in[12];
    const float* val_b = (const float*)d_in[13];
    const float* off_w = (const float*)d_in[14];
    const float* off_b = (const float*)d_in[15];
    const float* aw_w  = (const float*)d_in[16];
    const float* aw_b  = (const float*)d_in[17];
    const float* op_w  = (const float*)d_in[18];
    const float* op_b  = (const float*)d_in[19];
    const float* ln1_w = (const float*)d_in[20];
    const float* ln1_b = (const float*)d_in[21];
    const float* w1    = (const float*)d_in[22];
    const float* b1    = (const float*)d_in[23];
    const float* w2    = (const float*)d_in[24];
    const float* b2    = (const float*)d_in[25];
    const float* ln3_w = (const float*)d_in[26];
    const float* ln3_b = (const float*)d_in[27];
    float* out = (float*)d_out;

    float* ws = (float*)d_ws;
    const size_t NQf = (size_t)BB * LQ * DD;           // 1,843,200
    float* q_buf  = ws;                                //  NQf
    float* qh     = ws + 1 * NQf;                      //  NQf
    float* kh     = ws + 2 * NQf;                      //  NQf
    float* vh     = ws + 3 * NQf;                      //  NQf
    float* attno  = ws + 4 * NQf;                      //  NQf
    float* tgt2   = ws + 5 * NQf;                      //  NQf
    float* tgt3   = ws + 6 * NQf;                      //  NQf
    float* hidden = ws + 7 * NQf;                      //  4*NQf (7200x1024)
    float* value  = ws + 11 * NQf;                     //  B*LEN*D = 27,226,112
    // reuse:
    float* sa    = q_buf;    // q dead after qh/kh GEMMs
    float* q2    = qh;       // qh dead after attention
    float* offb  = kh;       // kh dead after attention
    float* awb   = vh;       // vh dead after attention
    float* capre = attno;    // attno dead after out-proj
    float* ca    = q_buf;    // sa dead after first LN
    float* ff    = qh;       // q2 dead after off/aw GEMMs

    const int MROWS  = BB * LQ;            // 7200 (= 450 * 16)
    const int MVAL   = BB * LEN;           // 106352 (= 6647 * 16)
    const int nEl    = (int)NQf;
    const int nWaves = BB * NH * LQ;       // 57600

    // ---- self attention ----
    add_vec<<<(nEl + 255) / 256, 256, 0, stream>>>(tgt, query_pos, q_buf, nEl);
    gemm(q_buf, qkv_w,        qkv_b,        qh, MROWS, DD, DD, 3 * DD, 0, stream);
    gemm(q_buf, qkv_w + DD,   qkv_b + DD,   kh, MROWS, DD, DD, 3 * DD, 0, stream);
    gemm(tgt,   qkv_w + 2*DD, qkv_b + 2*DD, vh, MROWS, DD, DD, 3 * DD, 0, stream);
    self_attn<<<(nWaves * 32 + 255) / 256, 256, 0, stream>>>(qh, kh, vh, attno);
    gemm(attno, out_w, out_b, sa, MROWS, DD, DD, DD, 0, stream);
    add_ln256<<<(MROWS * 32 + 255) / 256, 256, 0, stream>>>(tgt, sa, ln2_w, ln2_b, tgt2, MROWS);

    // ---- ms-deformable cross attention ----
    add_vec<<<(nEl + 255) / 256, 256, 0, stream>>>(tgt2, query_pos, q2, nEl);
    gemm(q2,  off_w, off_b, offb, MROWS, DD, DD,  DD,  0, stream);
    gemm(q2,  aw_w,  aw_b,  awb,  MROWS, DD, 128, 128, 0, stream);
    gemm(src, val_w, val_b, value, MVAL, DD, DD,  DD,  0, stream);
    msdeform_sample<<<(nWaves * 32 + 255) / 256, 256, 0, stream>>>(value, offb, awb, refpts, capre);
    gemm(capre, op_w, op_b, ca, MROWS, DD, DD, DD, 0, stream);
    add_ln256<<<(MROWS * 32 + 255) / 256, 256, 0, stream>>>(tgt2, ca, ln1_w, ln1_b, tgt3, MROWS);

    // ---- FFN ----
    gemm(tgt3,   w1, b1, hidden, MROWS, DD,   DFFN, DFFN, 1, stream);
    gemm(hidden, w2, b2, ff,     MROWS, DFFN, DD,   DD,   0, stream);
    add_ln256<<<(MROWS * 32 + 255) / 256, 256, 0, stream>>>(tgt3, ff, ln3_w, ln3_b, out, MROWS);
}